// HEATConv_63977832841972
// MI455X (gfx1250) — compile-verified
//
#include <hip/hip_runtime.h>
#include <hip/hip_bf16.h>

typedef __attribute__((ext_vector_type(2))) float    v2f;
typedef __attribute__((ext_vector_type(8))) float    v8f;
typedef __attribute__((ext_vector_type(4))) unsigned v4u;
typedef __attribute__((ext_vector_type(8))) int      v8i;
typedef __attribute__((ext_vector_type(4))) int      v4i;

#define HD   128
#define TN   8
#define TE_  16
#define ETE_ 32
#define EAE_ 32
#define ED_  16
#define GNUM 64
#define ODIM 64

__device__ __forceinline__ float lrelu(float v, float s) { return v >= 0.f ? v : s * v; }
__device__ __forceinline__ unsigned encf(float f) {
  unsigned u = __float_as_uint(f);
  return (u & 0x80000000u) ? ~u : (u | 0x80000000u);
}
__device__ __forceinline__ float decf(unsigned u) {
  return __uint_as_float((u & 0x80000000u) ? (u & 0x7fffffffu) : ~u);
}

// --------------------------------------------------------------- TDM 2D tile
// Build a D# (groups 0..3) for a 2D row-major f32 tensor [rows x cols] and DMA
// a [tile_rows x cols] tile starting at gbase into LDS at lds_off, with
// hardware LDS padding (pad_amount=1 DWORD every 2^(pad_code+1) DWORDs) so the
// LDS image has row stride cols+1 (bank-conflict-free A-fragment reads).
// OOB rows (>= rows) read as zero -> free masking of the ragged last tile.
// This toolchain's builtin is the 6-arg clang-23 form:
//   (uint32x4 g0, int32x8 g1, int32x4 g2, int32x4 g3, int32x8 extra, i32 cpol)
__device__ __forceinline__ void tdm_load_tile_2d(
    unsigned lds_off, const float* gbase, unsigned cols, unsigned rows,
    unsigned tile_rows, unsigned pad_code) {
  unsigned long long ga = (unsigned long long)(uintptr_t)gbase;
  v4u g0;
  g0[0] = 1u;                               // count=1, user descriptor
  g0[1] = lds_off;                          // LDS byte address
  g0[2] = (unsigned)ga;                     // global_addr[31:0]
  g0[3] = (unsigned)(ga >> 32) | (2u << 30);// global_addr[56:32] | type=2
  v8i g1;
  g1[0] = (int)((2u << 16) | (1u << 20) | (pad_code << 22)); // 4B elems, pad_en, interval
  g1[1] = (int)((cols & 0xFFFFu) << 16);                     // tensor_dim0 lo
  g1[2] = (int)((cols >> 16) | ((rows & 0xFFFFu) << 16));    // dim0 hi | tensor_dim1 lo
  g1[3] = (int)((rows >> 16) | (cols << 16));                // dim1 hi | tile_dim0=cols
  g1[4] = (int)tile_rows;                                    // tile_dim1, tile_dim2=0
  g1[5] = (int)cols;                                         // tensor_dim0_stride lo
  g1[6] = 0;                                                 // stride hi | dim1_stride lo
  g1[7] = 0;
  v4i g2 = {0, 0, 0, 0};                                     // 2D: dims2/3 unused
  v4i g3 = {0, 0, 0, 0};
  v8i gx = {0, 0, 0, 0, 0, 0, 0, 0};
  __builtin_amdgcn_tensor_load_to_lds(g0, g1, g2, g3, gx, 0);
}

// ---------------------------------------------------------------- fill
__global__ void fill_kernel(unsigned* __restrict__ p, unsigned v, int n) {
  int i = blockIdx.x * blockDim.x + threadIdx.x;
  if (i < n) p[i] = v;
}

// ------------------------------------------- HeteroLinear via WMMA f32 16x16x4
// 16-node tile DMA'd to LDS by the TDM; 4 waves each own 32 output columns.
// For each node-type present, run the K=128 WMMA chain and store rows whose
// type matches (row types held in per-lane VGPRs, -1 = OOB row).
__global__ __launch_bounds__(128) void hetero_kernel(
    const float* __restrict__ xin, const float* __restrict__ W,
    const float* __restrict__ bias, const int* __restrict__ ntype,
    float* __restrict__ out, int n, int relu_in) {
  __shared__ float xs[16][HD + 1];
  __shared__ int trow[16];
  __shared__ unsigned tmask;
  const int tid = threadIdx.x;
  const int tile0 = blockIdx.x * 16;
  if (tid == 0) tmask = 0u;
  if (tid < 32) {  // wave 0 issues the DMA (TDM ignores EXEC; once per wave)
    tdm_load_tile_2d((unsigned)(uintptr_t)&xs[0][0], xin + (size_t)tile0 * HD,
                     HD, (unsigned)n, 16, 6u /* pad every 128 dwords */);
    __builtin_amdgcn_s_wait_tensorcnt(0);
  }
  if (tid < 16) {
    int node = tile0 + tid;
    int t = (node < n) ? ntype[node] : -1;
    trow[tid] = t;
    if (node < n) atomicOr(&tmask, 1u << t);
  }
  __syncthreads();
  if (relu_in) {
    for (int i = 0; i < 16; ++i) {
      float v = xs[i][tid];
      xs[i][tid] = v > 0.f ? v : 0.f;
    }
    __syncthreads();
  }
  const int lane = tid & 31;
  const int wv = tid >> 5;
  const int arow = lane & 15;
  const unsigned mask = tmask;
  int rowt[8];
  #pragma unroll
  for (int v = 0; v < 8; ++v) rowt[v] = trow[(lane < 16) ? v : v + 8];
  for (int t = 0; t < TN; ++t) {
    if (!((mask >> t) & 1u)) continue;
    const float* Wt = W + t * HD * HD;
    for (int ntile = 0; ntile < 2; ++ntile) {
      const int cb = wv * 32 + ntile * 16 + (lane & 15);
      v8f c = {0.f, 0.f, 0.f, 0.f, 0.f, 0.f, 0.f, 0.f};
      #pragma unroll 4
      for (int kt = 0; kt < HD / 4; ++kt) {
        const int ak = kt * 4 + ((lane >> 4) << 1);  // lanes 0-15: k0,k0+1; 16-31: k0+2,k0+3
        v2f a, b;
        a.x = xs[arow][ak];
        a.y = xs[arow][ak + 1];
        b.x = Wt[ak * HD + cb];
        b.y = Wt[(ak + 1) * HD + cb];
        c = __builtin_amdgcn_wmma_f32_16x16x4_f32(false, a, false, b, (short)0, c, false, false);
      }
      const float bb = bias[t * HD + cb];
      #pragma unroll
      for (int v = 0; v < 8; ++v) {
        const int m = (lane < 16) ? v : v + 8;
        if (rowt[v] == t) out[(tile0 + m) * HD + cb] = c[v] + bb;
      }
    }
  }
}

// ------------------------------------------- node GEMM y = h @ lin_W[:128,:]
__global__ __launch_bounds__(128) void nodegemm_kernel(
    const float* __restrict__ xin, const float* __restrict__ W,
    float* __restrict__ out, int n) {
  __shared__ float xs[16][HD + 1];
  const int tid = threadIdx.x;
  const int tile0 = blockIdx.x * 16;
  if (tid < 32) {
    tdm_load_tile_2d((unsigned)(uintptr_t)&xs[0][0], xin + (size_t)tile0 * HD,
                     HD, (unsigned)n, 16, 6u);
    __builtin_amdgcn_s_wait_tensorcnt(0);
  }
  __syncthreads();
  const int lane = tid & 31;
  const int wv = tid >> 5;
  const int arow = lane & 15;
  for (int ntile = 0; ntile < 2; ++ntile) {
    const int cb = wv * 32 + ntile * 16 + (lane & 15);
    v8f c = {0.f, 0.f, 0.f, 0.f, 0.f, 0.f, 0.f, 0.f};
    #pragma unroll 4
    for (int kt = 0; kt < HD / 4; ++kt) {
      const int ak = kt * 4 + ((lane >> 4) << 1);
      v2f a, b;
      a.x = xs[arow][ak];
      a.y = xs[arow][ak + 1];
      b.x = W[ak * HD + cb];
      b.y = W[(ak + 1) * HD + cb];
      c = __builtin_amdgcn_wmma_f32_16x16x4_f32(false, a, false, b, (short)0, c, false, false);
    }
    #pragma unroll
    for (int v = 0; v < 8; ++v) {
      const int m = (lane < 16) ? v : v + 8;
      const int node = tile0 + m;
      if (node < n) out[node * HD + cb] = c[v];
    }
  }
}

// ------------------------------------------- per-node attention scalars
// a_i[n] = h[n]·attW[0:128], a_j[n] = h[n]·attW[128:256]; one wave per node.
__global__ __launch_bounds__(256) void node_dots_kernel(
    const float* __restrict__ h, const float* __restrict__ attW,
    float* __restrict__ a_i, float* __restrict__ a_j, int n) {
  const int lane = threadIdx.x & 31;
  const int node = blockIdx.x * 8 + (threadIdx.x >> 5);
  if (node >= n) return;
  const float4 xv = *(const float4*)(h + node * HD + lane * 4);
  const float4 wi = *(const float4*)(attW + lane * 4);
  const float4 wj = *(const float4*)(attW + HD + lane * 4);
  float si = xv.x * wi.x + xv.y * wi.y + xv.z * wi.z + xv.w * wi.w;
  float sj = xv.x * wj.x + xv.y * wj.y + xv.z * wj.z + xv.w * wj.w;
  for (int off = 16; off > 0; off >>= 1) {
    si += __shfl_down(si, off, 32);
    sj += __shfl_down(sj, off, 32);
  }
  if (lane == 0) { a_i[node] = si; a_j[node] = sj; }
}

// ------------------------------------------- per-edge-type attention scalar
__global__ void ate_kernel(const float* __restrict__ ete, const float* __restrict__ attW,
                           float* __restrict__ a_te) {
  int t = threadIdx.x;
  if (t < TE_) {
    float s = 0.f;
    for (int j = 0; j < ETE_; ++j) s += lrelu(ete[t * ETE_ + j], 0.01f) * attW[2 * HD + j];
    a_te[t] = s;
  }
}

// ------------------------------------------- pass A: alpha + segment max
__global__ __launch_bounds__(256) void alpha_kernel(
    const int* __restrict__ src, const int* __restrict__ dst, const int* __restrict__ etype,
    const float* __restrict__ eattr, const float* __restrict__ eaeW,
    const float* __restrict__ attW, const float* __restrict__ a_i,
    const float* __restrict__ a_j, const float* __restrict__ a_te,
    float* __restrict__ alpha, unsigned* __restrict__ amaxu, int E_) {
  int e = blockIdx.x * 256 + threadIdx.x;
  if (e >= E_) return;
  float ar[ED_];
  #pragma unroll
  for (int i = 0; i < ED_; ++i) ar[i] = eattr[e * ED_ + i];
  float a_eae = 0.f;
  #pragma unroll 4
  for (int j = 0; j < EAE_; ++j) {
    float s = 0.f;
    #pragma unroll
    for (int i = 0; i < ED_; ++i) s += ar[i] * eaeW[i * EAE_ + j];
    a_eae += lrelu(s, 0.01f) * attW[2 * HD + ETE_ + j];
  }
  const int si = src[e], di = dst[e];
  float al = a_i[di] + a_j[si] + a_te[etype[e]] + a_eae;
  al = lrelu(al, 0.2f);
  alpha[e] = al;
  atomicMax(amaxu + di, encf(al));
}

// ------------------------------------------- pass B: exp + segment sum
__global__ __launch_bounds__(256) void exp_kernel(
    const int* __restrict__ dst, const float* __restrict__ alpha,
    const unsigned* __restrict__ amaxu, float* __restrict__ exb,
    float* __restrict__ denom, int E_) {
  int e = blockIdx.x * 256 + threadIdx.x;
  if (e >= E_) return;
  const int d = dst[e];
  const float ex = __expf(alpha[e] - decf(amaxu[d]));
  exb[e] = ex;
  atomicAdd(denom + d, ex);
}

// ------------------------------------------- pass C: message + scatter-add
// 16-edge tile: TDM DMA of the edge_attr tile, recompute eae (K=16, cheap),
// WMMA z = eae @ lin_W[128:160,:], then (z + y[src] + b)*coef scattered into
// hout[dst] via f32 atomics (hot in the 192MB L2).
__global__ __launch_bounds__(128) void message_kernel(
    const int* __restrict__ src, const int* __restrict__ dst,
    const float* __restrict__ eattr, const float* __restrict__ eaeW,
    const float* __restrict__ linW, const float* __restrict__ linb,
    const float* __restrict__ y, const float* __restrict__ exb,
    const float* __restrict__ denom, float* __restrict__ hout, int E_) {
  __shared__ float ea[16][ED_ + 1];
  __shared__ float eaes[16][EAE_ + 1];
  __shared__ int s_src[16], s_dst[16];
  __shared__ float s_coef[16];
  const int tid = threadIdx.x;
  const int tile0 = blockIdx.x * 16;
  if (tid < 32) {
    tdm_load_tile_2d((unsigned)(uintptr_t)&ea[0][0], eattr + (size_t)tile0 * ED_,
                     ED_, (unsigned)E_, 16, 3u /* pad every 16 dwords */);
    __builtin_amdgcn_s_wait_tensorcnt(0);
  }
  if (tid < 16) {
    int e = tile0 + tid;
    if (e < E_) {
      int d = dst[e];
      s_src[tid] = src[e];
      s_dst[tid] = d;
      s_coef[tid] = exb[e] / denom[d];
    } else { s_src[tid] = 0; s_dst[tid] = 0; s_coef[tid] = 0.f; }
  }
  __syncthreads();
  #pragma unroll
  for (int rep = 0; rep < 4; ++rep) {
    int f = tid + rep * 128;
    int r = f >> 5, j = f & 31;
    float s = 0.f;
    #pragma unroll
    for (int k = 0; k < ED_; ++k) s += ea[r][k] * eaeW[k * EAE_ + j];
    eaes[r][j] = lrelu(s, 0.01f);
  }
  __syncthreads();
  const int lane = tid & 31;
  const int wv = tid >> 5;
  const int arow = lane & 15;
  const float* Wz = linW + HD * HD;  // rows 128..159 of lin_W
  for (int ntile = 0; ntile < 2; ++ntile) {
    const int cb = wv * 32 + ntile * 16 + (lane & 15);
    v8f c = {0.f, 0.f, 0.f, 0.f, 0.f, 0.f, 0.f, 0.f};
    #pragma unroll
    for (int kt = 0; kt < EAE_ / 4; ++kt) {
      const int ak = kt * 4 + ((lane >> 4) << 1);
      v2f a, b;
      a.x = eaes[arow][ak];
      a.y = eaes[arow][ak + 1];
      b.x = Wz[ak * HD + cb];
      b.y = Wz[(ak + 1) * HD + cb];
      c = __builtin_amdgcn_wmma_f32_16x16x4_f32(false, a, false, b, (short)0, c, false, false);
    }
    const float bb = linb[cb];
    #pragma unroll
    for (int v = 0; v < 8; ++v) {
      const int m = (lane < 16) ? v : v + 8;
      const int e = tile0 + m;
      if (e < E_) {
        float val = (c[v] + y[s_src[m] * HD + cb] + bb) * s_coef[m];
        atomicAdd(hout + s_dst[m] * HD + cb, val);
      }
    }
  }
}

// ------------------------------------------- global mean pool (sum + count)
__global__ __launch_bounds__(256) void pool_kernel(
    const float* __restrict__ h, const int* __restrict__ batch,
    float* __restrict__ pooled, float* __restrict__ cnt, int n) {
  const int lane = threadIdx.x & 31;
  const int node = blockIdx.x * 8 + (threadIdx.x >> 5);
  if (node >= n) return;
  const int g = batch[node];
  float4 v = *(const float4*)(h + node * HD + lane * 4);
  atomicAdd(pooled + g * HD + lane * 4 + 0, v.x > 0.f ? v.x : 0.f);
  atomicAdd(pooled + g * HD + lane * 4 + 1, v.y > 0.f ? v.y : 0.f);
  atomicAdd(pooled + g * HD + lane * 4 + 2, v.z > 0.f ? v.z : 0.f);
  atomicAdd(pooled + g * HD + lane * 4 + 3, v.w > 0.f ? v.w : 0.f);
  if (lane == 0) atomicAdd(cnt + g, 1.0f);
}

// ------------------------------------------- post MLP (tiny, one block/graph)
__global__ __launch_bounds__(128) void post_kernel(
    const float* __restrict__ pooled, const float* __restrict__ cnt,
    const float* __restrict__ W1, const float* __restrict__ b1,
    const float* __restrict__ W2, const float* __restrict__ b2,
    float* __restrict__ out) {
  __shared__ float p[HD];
  __shared__ float t1[HD];
  const int g = blockIdx.x;
  const int tid = threadIdx.x;
  float c = cnt[g];
  c = c > 1.0f ? c : 1.0f;
  p[tid] = pooled[g * HD + tid] / c;
  __syncthreads();
  float s = b1[tid];
  for (int d = 0; d < HD; ++d) s += p[d] * W1[d * HD + tid];
  t1[tid] = s;
  __syncthreads();
  if (tid < ODIM) {
    float o = b2[tid];
    for (int d = 0; d < HD; ++d) o += t1[d] * W2[d * ODIM + tid];
    out[g * ODIM + tid] = o;
  }
}

// ================================================================ launch
extern "C" void kernel_launch(void* const* d_in, const int* in_sizes, int n_in,
                              void* d_out, int out_size, void* d_ws, size_t ws_size,
                              hipStream_t stream) {
  const int N = in_sizes[2];   // node_type count
  const int E = in_sizes[3];   // edge_type count
  const float* x     = (const float*)d_in[0];
  const int* ei      = (const int*)d_in[1];
  const int* ntype   = (const int*)d_in[2];
  const int* etype   = (const int*)d_in[3];
  const float* eattr = (const float*)d_in[4];
  const int* batch   = (const int*)d_in[5];
  const int* src = ei;
  const int* dst = ei + E;

  // workspace layout (floats)
  float* hlin  = (float*)d_ws;          // N*HD
  float* ybuf  = hlin + (size_t)N * HD; // N*HD
  float* hbuf  = ybuf + (size_t)N * HD; // N*HD (layer output accumulator)
  float* a_i   = hbuf + (size_t)N * HD; // N
  float* a_j   = a_i + N;               // N
  unsigned* amaxu = (unsigned*)(a_j + N);   // N
  float* denom = (float*)(amaxu + N);   // N
  float* a_te  = denom + N;             // TE_
  float* alpha = a_te + TE_;            // E
  float* exb   = alpha + E;             // E
  float* pooled = exb + E;              // GNUM*HD
  float* cnt   = pooled + GNUM * HD;    // GNUM

  const int nodeTiles = (N + 15) / 16;
  const int edgeTiles = (E + 15) / 16;
  const int nodeWaveBlocks = (N + 7) / 8;

  for (int l = 0; l < 2; ++l) {
    const float* hW   = (const float*)d_in[6 + 7 * l + 0];
    const float* hb   = (const float*)d_in[6 + 7 * l + 1];
    const float* ete  = (const float*)d_in[6 + 7 * l + 2];
    const float* eaeW = (const float*)d_in[6 + 7 * l + 3];
    const float* attW = (const float*)d_in[6 + 7 * l + 4];
    const float* linW = (const float*)d_in[6 + 7 * l + 5];
    const float* linb = (const float*)d_in[6 + 7 * l + 6];
    const float* hin = (l == 0) ? x : hbuf;

    hetero_kernel<<<nodeTiles, 128, 0, stream>>>(hin, hW, hb, ntype, hlin, N, l > 0);
    node_dots_kernel<<<nodeWaveBlocks, 256, 0, stream>>>(hlin, attW, a_i, a_j, N);
    ate_kernel<<<1, 32, 0, stream>>>(ete, attW, a_te);
    nodegemm_kernel<<<nodeTiles, 128, 0, stream>>>(hlin, linW, ybuf, N);

    fill_kernel<<<(N + 255) / 256, 256, 0, stream>>>(amaxu, 0u, N);
    fill_kernel<<<(N + 255) / 256, 256, 0, stream>>>((unsigned*)denom, 0u, N);
    fill_kernel<<<(N * HD + 255) / 256, 256, 0, stream>>>((unsigned*)hbuf, 0u, N * HD);

    alpha_kernel<<<(E + 255) / 256, 256, 0, stream>>>(src, dst, etype, eattr, eaeW, attW,
                                                      a_i, a_j, a_te, alpha, amaxu, E);
    exp_kernel<<<(E + 255) / 256, 256, 0, stream>>>(dst, alpha, amaxu, exb, denom, E);
    message_kernel<<<edgeTiles, 128, 0, stream>>>(src, dst, eattr, eaeW, linW, linb,
                                                  ybuf, exb, denom, hbuf, E);
  }

  const float* W1 = (const float*)d_in[20];
  const float* b1 = (const float*)d_in[21];
  const float* W2 = (const float*)d_in[22];
  const float* b2 = (const float*)d_in[23];

  fill_kernel<<<(GNUM * HD + 255) / 256, 256, 0, stream>>>((unsigned*)pooled, 0u, GNUM * HD);
  fill_kernel<<<1, 64, 0, stream>>>((unsigned*)cnt, 0u, GNUM);
  pool_kernel<<<nodeWaveBlocks, 256, 0, stream>>>(hbuf, batch, pooled, cnt, N);
  post_kernel<<<GNUM, 128, 0, stream>>>(pooled, cnt, W1, b1, W2, b2, (float*)d_out);
}